// YoloXLoss_7121055777339
// MI455X (gfx1250) — compile-verified
//
#include <hip/hip_runtime.h>
#include <float.h>
#include <math.h>

#define NB   64
#define NA   6300
#define NCH  85
#define NT   1024
#define NCLS 80
#define INV_PI2_4 0.4052847345693511f   /* 4 / pi^2 */

typedef float v2f __attribute__((ext_vector_type(2)));
typedef float v8f __attribute__((ext_vector_type(8)));

__device__ __forceinline__ void anchor_xy(int a, float& ax, float& ay, float& s) {
  if (a < 4800) {
    s = 8.0f;  int x = a % 80;  int y = a / 80;
    ax = (x + 0.5f) * 8.0f;  ay = (y + 0.5f) * 8.0f;
  } else if (a < 6000) {
    s = 16.0f; int r = a - 4800; int x = r % 40; int y = r / 40;
    ax = (x + 0.5f) * 16.0f; ay = (y + 0.5f) * 16.0f;
  } else {
    s = 32.0f; int r = a - 6000; int x = r % 20; int y = r / 20;
    ax = (x + 0.5f) * 32.0f; ay = (y + 0.5f) * 32.0f;
  }
}

__device__ __forceinline__ float bce_logits(float x, float t) {
  return fmaxf(x, 0.0f) - x * t + log1pf(expf(-fabsf(x)));
}

// ---------------------------------------------------------------- init
__global__ void k_init(int* matched, float* obj_sum, float* box_sum,
                       float* cls_sum, float* nfg, int* first_t) {
  int gid = blockIdx.x * blockDim.x + threadIdx.x;
  if (gid < NB * NA) matched[gid] = -1;
  if (gid == 0) obj_sum[0] = 0.0f;
  if (gid < NB) { box_sum[gid] = 0.0f; cls_sum[gid] = 0.0f; nfg[gid] = 0.0f; first_t[gid] = NT; }
}

// ------------------------------------------------- matching via WMMA
// d2[t][a] = |g|^2 + |a|^2 - 2 g.a  as a K=4 inner product:
//   A row t = [gx, gy, |g|^2, 1],  B col a = [-2ax, -2ay, 1, |a|^2]
// One V_WMMA_F32_16X16X4_F32 yields a full 16x16 d2 tile.
__global__ __launch_bounds__(256) void k_match(const float* __restrict__ targets,
                                               int* __restrict__ matched) {
  __shared__ float s_dist[16 * 128];
  __shared__ float s_md[16][16][10];
  __shared__ int   s_mi[16][16][10];

  const int tid   = threadIdx.x;
  const int lane  = tid & 31;
  const int wv    = tid >> 5;           // 8 waves
  const int tbase = blockIdx.x * 16;    // 16 targets per block

  // A operand (16x4 f32): lanes 0-15 hold K0,K1; lanes 16-31 hold K2,K3
  const int   m  = lane & 15;
  const float gx = targets[(tbase + m) * 6 + 2] * 640.0f;
  const float gy = targets[(tbase + m) * 6 + 3] * 480.0f;
  v2f Aop;
  Aop[0] = (lane < 16) ? gx : (gx * gx + gy * gy);
  Aop[1] = (lane < 16) ? gy : 1.0f;

  float bd[10]; int bi[10];
  #pragma unroll
  for (int i = 0; i < 10; ++i) { bd[i] = FLT_MAX; bi[i] = 0x7fffffff; }

  const int row  = tid >> 4;   // target row 0..15
  const int slot = tid & 15;   // 16 reducer threads per row

  for (int chunk = 0; chunk < 50; ++chunk) {        // 50*128 = 6400 >= 6300
    const int  ai  = chunk * 128 + wv * 16 + (lane & 15);
    const bool pad = (ai >= NA);
    float ax = 0.0f, ay = 0.0f, ss = 8.0f;
    if (!pad) anchor_xy(ai, ax, ay, ss);
    v2f Bop;
    Bop[0] = (lane < 16) ? (-2.0f * ax) : 1.0f;
    Bop[1] = (lane < 16) ? (-2.0f * ay) : (pad ? 1e30f : (ax * ax + ay * ay));

    v8f C = {};
    v8f D = __builtin_amdgcn_wmma_f32_16x16x4_f32(false, Aop, false, Bop,
                                                  (short)0, C, false, false);
    // D layout: lanes 0-15 -> rows 0..7 in D[0..7]; lanes 16-31 -> rows 8..15
    const int col = wv * 16 + (lane & 15);
    const int ro  = (lane < 16) ? 0 : 8;
    #pragma unroll
    for (int i = 0; i < 8; ++i) s_dist[(ro + i) * 128 + col] = D[i];
    __syncthreads();

    #pragma unroll
    for (int j = 0; j < 8; ++j) {
      const int   c    = slot + j * 16;
      const int   aidx = chunk * 128 + c;
      const float dd   = sqrtf(fmaxf(s_dist[row * 128 + c], 0.0f));
      bool ins = (dd < bd[9]) || (dd == bd[9] && aidx < bi[9]);
      if (ins) {
        bd[9] = dd; bi[9] = aidx;
        #pragma unroll
        for (int k = 9; k > 0; --k) {
          bool sw = (bd[k] < bd[k-1]) || (bd[k] == bd[k-1] && bi[k] < bi[k-1]);
          if (sw) {
            float td = bd[k]; bd[k] = bd[k-1]; bd[k-1] = td;
            int   ti = bi[k]; bi[k] = bi[k-1]; bi[k-1] = ti;
          }
        }
      }
    }
    __syncthreads();
  }

  #pragma unroll
  for (int i = 0; i < 10; ++i) { s_md[row][slot][i] = bd[i]; s_mi[row][slot][i] = bi[i]; }
  __syncthreads();

  if (slot == 0) {   // 16-way merge of sorted 10-lists -> global top-10, scatter
    int ptrs[16];
    #pragma unroll
    for (int s2 = 0; s2 < 16; ++s2) ptrs[s2] = 0;
    const int tg   = tbase + row;
    const int timg = (int)targets[tg * 6 + 0];
    const int base = timg * NA;
    for (int k = 0; k < 10; ++k) {
      float best = FLT_MAX; int besti = 0x7fffffff; int bs = 0;
      #pragma unroll
      for (int s2 = 0; s2 < 16; ++s2) {
        int   p  = ptrs[s2];
        float dv = s_md[row][s2][p];
        int   iv = s_mi[row][s2][p];
        if (dv < best || (dv == best && iv < besti)) { best = dv; besti = iv; bs = s2; }
      }
      ptrs[bs]++;
      atomicMax(&matched[base + besti], tg);
    }
  }
}

// ----------------------------------------------------------- first_t
__global__ void k_first(const float* __restrict__ targets, int* __restrict__ first_t) {
  int t = blockIdx.x * blockDim.x + threadIdx.x;
  if (t < NT) {
    int timg = (int)targets[t * 6 + 0];
    atomicMin(&first_t[timg], t);
  }
}

// ---------------------------------------------------------- main loss
__global__ __launch_bounds__(256) void k_loss(
    const float* __restrict__ out0, const float* __restrict__ out1,
    const float* __restrict__ out2, const float* __restrict__ targets,
    const int* __restrict__ matched, const int* __restrict__ first_t,
    float* __restrict__ obj_sum, float* __restrict__ box_sum,
    float* __restrict__ cls_sum, float* __restrict__ nfg) {
  __shared__ float red[256];
  const int gid = blockIdx.x * blockDim.x + threadIdx.x;
  float obj = 0.0f;
  if (gid < NB * NA) {
    const int b = gid / NA;
    const int a = gid - b * NA;
    const float* p;
    if (a < 4800)      p = out0 + ((size_t)b * 4800 + a) * NCH;
    else if (a < 6000) p = out1 + ((size_t)b * 1200 + (a - 4800)) * NCH;
    else               p = out2 + ((size_t)b * 300  + (a - 6000)) * NCH;

    const int  mt  = matched[gid];
    const float fgf = (mt >= 0) ? 1.0f : 0.0f;
    obj = bce_logits(p[4], fgf);

    if (mt >= 0) {
      const float eps = 1e-7f;
      float ax, ay, s; anchor_xy(a, ax, ay, s);
      float o0 = p[0], o1 = p[1], o2 = p[2], o3 = p[3];
      float pcx = (1.0f / (1.0f + expf(-o0))) * s + ax;
      float pcy = (1.0f / (1.0f + expf(-o1))) * s + ay;
      float pw  = expf(fminf(o2, 4.0f)) * s;
      float ph  = expf(fminf(o3, 4.0f)) * s;
      float gcx = targets[mt * 6 + 2] * 640.0f;
      float gcy = targets[mt * 6 + 3] * 480.0f;
      float gw  = targets[mt * 6 + 4] * 640.0f;
      float gh  = targets[mt * 6 + 5] * 480.0f;
      float px1 = pcx - pw * 0.5f, py1 = pcy - ph * 0.5f;
      float px2 = pcx + pw * 0.5f, py2 = pcy + ph * 0.5f;
      float qx1 = gcx - gw * 0.5f, qy1 = gcy - gh * 0.5f;
      float qx2 = gcx + gw * 0.5f, qy2 = gcy + gh * 0.5f;
      float iw = fmaxf(fminf(px2, qx2) - fmaxf(px1, qx1), 0.0f);
      float ih = fmaxf(fminf(py2, qy2) - fmaxf(py1, qy1), 0.0f);
      float inter = iw * ih;
      float uni = (px2 - px1) * (py2 - py1) + (qx2 - qx1) * (qy2 - qy1) - inter;
      float iou = inter / (uni + eps);
      float cxd = (pcx - gcx) * (pcx - gcx) + (pcy - gcy) * (pcy - gcy);
      float dgx = fmaxf(px2, qx2) - fminf(px1, qx1);
      float dgy = fmaxf(py2, qy2) - fminf(py1, qy1);
      float diag = dgx * dgx + dgy * dgy + eps;
      float dat = atanf(gw / (gh + eps)) - atanf(pw / (ph + eps));
      float v = INV_PI2_4 * dat * dat;
      float alpha = v / (1.0f - iou + v + eps);
      float lbox = 1.0f - iou + cxd / diag + alpha * v;
      atomicAdd(&box_sum[b], lbox);

      int ft = first_t[b];
      int ti = (ft < NT) ? ft : (NT - 1);
      int ci = (int)targets[ti * 6 + 1];
      float csum = 0.0f;
      #pragma unroll 4
      for (int c = 0; c < NCLS; ++c) {
        float tt = (c == ci) ? 1.0f : 0.0f;
        csum += bce_logits(p[5 + c], tt);
      }
      atomicAdd(&cls_sum[b], csum);
      atomicAdd(&nfg[b], 1.0f);
    }
  }
  // block reduction of obj BCE
  red[threadIdx.x] = obj;
  __syncthreads();
  for (int off = 128; off > 0; off >>= 1) {
    if (threadIdx.x < off) red[threadIdx.x] += red[threadIdx.x + off];
    __syncthreads();
  }
  if (threadIdx.x == 0) atomicAdd(obj_sum, red[0]);
}

// ----------------------------------------------------------- finalize
__global__ void k_final(const float* obj_sum, const float* box_sum,
                        const float* cls_sum, const float* nfg, float* out) {
  if (threadIdx.x == 0 && blockIdx.x == 0) {
    float lobj = obj_sum[0] / (float)(NB * NA);
    float lbox = 0.0f, lcls = 0.0f, nt = 0.0f;
    for (int b = 0; b < NB; ++b) {
      float d = fmaxf(nfg[b], 1.0f);
      lbox += box_sum[b] / d;
      lcls += cls_sum[b] / (d * (float)NCLS);
      nt   += nfg[b];
    }
    float norm = fmaxf(1.0f, nt / (float)NB);
    out[0] = lobj + lcls / norm + 5.0f * lbox / norm;
  }
}

extern "C" void kernel_launch(void* const* d_in, const int* in_sizes, int n_in,
                              void* d_out, int out_size, void* d_ws, size_t ws_size,
                              hipStream_t stream) {
  const float* out0    = (const float*)d_in[0];
  const float* out1    = (const float*)d_in[1];
  const float* out2    = (const float*)d_in[2];
  const float* targets = (const float*)d_in[3];

  int*   matched = (int*)d_ws;                    // NB*NA ints  (~1.6 MB)
  float* obj_sum = (float*)(matched + NB * NA);   // 1
  float* box_sum = obj_sum + 1;                   // NB
  float* cls_sum = box_sum + NB;                  // NB
  float* nfg     = cls_sum + NB;                  // NB
  int*   first_t = (int*)(nfg + NB);              // NB

  const int nblocks = (NB * NA + 255) / 256;      // 1575, exact
  hipLaunchKernelGGL(k_init,  dim3(nblocks), dim3(256), 0, stream,
                     matched, obj_sum, box_sum, cls_sum, nfg, first_t);
  hipLaunchKernelGGL(k_match, dim3(NT / 16), dim3(256), 0, stream, targets, matched);
  hipLaunchKernelGGL(k_first, dim3((NT + 255) / 256), dim3(256), 0, stream, targets, first_t);
  hipLaunchKernelGGL(k_loss,  dim3(nblocks), dim3(256), 0, stream,
                     out0, out1, out2, targets, matched, first_t,
                     obj_sum, box_sum, cls_sum, nfg);
  hipLaunchKernelGGL(k_final, dim3(1), dim3(1), 0, stream,
                     obj_sum, box_sum, cls_sum, nfg, (float*)d_out);
}